// NIQE_69475390980608
// MI455X (gfx1250) — compile-verified
//
#include <hip/hip_runtime.h>
#include <math.h>
#include <stdint.h>

#define BATCH 16
#define H1 1056
#define W1 1056
#define H2 528
#define W2 528
#define NBW 11
#define NB 121          // blocks per image per scale (11x11 at both scales)
#define NF 36
#define NG 9801         // AGGD gamma-grid entries: arange(0.2, 10.001, 0.001)
#define KPAD 124        // NB padded to multiple of 4  (WMMA K)
#define FPAD 48         // NF padded to multiple of 16 (WMMA M/N)

typedef __attribute__((ext_vector_type(2))) float v2f;
typedef __attribute__((ext_vector_type(8))) float v8f;

// ---------------------------------------------------------------------------
// r_gam[i] = exp(2*lgamma(2/a) - lgamma(1/a) - lgamma(3/a)),  a = 0.2 + 0.001*i
// ---------------------------------------------------------------------------
__global__ void rgam_init_kernel(float* __restrict__ rgam) {
  int i = blockIdx.x * blockDim.x + threadIdx.x;
  if (i < NG) {
    float a = 0.2f + 0.001f * (float)i;
    rgam[i] = expf(2.f * lgammaf(2.f / a) - lgammaf(1.f / a) - lgammaf(3.f / a));
  }
}

// ---------------------------------------------------------------------------
// Fused separable 7x7 gaussian (replicate pad) + MSCN normalize.
// 16x16 outputs per workgroup, halo 3, LDS-resident.  sigma = 7/6.
// ---------------------------------------------------------------------------
__global__ __launch_bounds__(256) void mscn_kernel(const float* __restrict__ src,
                                                   float* __restrict__ dst,
                                                   int H, int W) {
  __shared__ float t[22][22];
  __shared__ float hv[22][16];
  __shared__ float hv2[22][16];
  const int tid = threadIdx.x;
  const int b = blockIdx.z;
  const int ox0 = blockIdx.x * 16 - 3;
  const int oy0 = blockIdx.y * 16 - 3;
  const float* sp = src + (size_t)b * H * W;

  float wgt[7];
  {
    float s = 0.f;
#pragma unroll
    for (int d = 0; d < 7; ++d) {
      int r = d - 3;
      wgt[d] = expf(-(float)(r * r) * (18.f / 49.f));  // 1/(2*(7/6)^2) = 18/49
      s += wgt[d];
    }
#pragma unroll
    for (int d = 0; d < 7; ++d) wgt[d] /= s;
  }

  for (int idx = tid; idx < 22 * 22; idx += 256) {
    int ly = idx / 22, lx = idx % 22;
    int y = oy0 + ly; y = y < 0 ? 0 : (y >= H ? H - 1 : y);
    int x = ox0 + lx; x = x < 0 ? 0 : (x >= W ? W - 1 : x);
    t[ly][lx] = sp[(size_t)y * W + x];
  }
  {  // speculative prefetch of the next tile row (gfx1250 global_prefetch_b8)
    int py = oy0 + 22; py = py < 0 ? 0 : (py >= H ? H - 1 : py);
    int px = ox0 + 3 + (tid & 15); px = px < 0 ? 0 : (px >= W ? W - 1 : px);
    __builtin_prefetch(&sp[(size_t)py * W + px], 0, 1);
  }
  __syncthreads();

  for (int idx = tid; idx < 22 * 16; idx += 256) {
    int ly = idx / 16, lx = idx % 16;
    float s = 0.f, s2 = 0.f;
#pragma unroll
    for (int d = 0; d < 7; ++d) {
      float v = t[ly][lx + d];
      s = fmaf(wgt[d], v, s);
      s2 = fmaf(wgt[d], v * v, s2);
    }
    hv[ly][lx] = s;
    hv2[ly][lx] = s2;
  }
  __syncthreads();

  const int ty = tid / 16, tx = tid % 16;
  float mu = 0.f, m2 = 0.f;
#pragma unroll
  for (int d = 0; d < 7; ++d) {
    mu = fmaf(wgt[d], hv[ty + d][tx], mu);
    m2 = fmaf(wgt[d], hv2[ty + d][tx], m2);
  }
  float sig = sqrtf(fabsf(m2 - mu * mu));
  float v = t[ty + 3][tx + 3];
  int y = blockIdx.y * 16 + ty, x = blockIdx.x * 16 + tx;
  dst[((size_t)b * H + y) * W + x] = (v - mu) / (sig + 1.f);
}

// ---------------------------------------------------------------------------
// 2x2 average-pool downsample 1056 -> 528
// ---------------------------------------------------------------------------
__global__ void downsample_kernel(const float* __restrict__ src, float* __restrict__ dst) {
  int i = blockIdx.x * blockDim.x + threadIdx.x;
  const int total = BATCH * H2 * W2;
  if (i >= total) return;
  int x = i % W2, y = (i / W2) % H2, b = i / (W2 * H2);
  const float* s = src + ((size_t)b * H1 + 2 * y) * W1 + 2 * x;
  dst[i] = 0.25f * (s[0] + s[1] + s[W1] + s[W1 + 1]);
}

// ---------------------------------------------------------------------------
// Per-block AGGD features (18 per scale).  One workgroup per (block, batch).
// Block pixels staged into LDS with gfx1250 async global->LDS B128 copies
// (ASYNCcnt path, no VGPR staging); 5 products x 5 stats reduced with wave32
// shuffles; 9801-point argmin via packed (|d|-bits<<32 | idx) min-reduce.
// ---------------------------------------------------------------------------
__global__ __launch_bounds__(256) void feat_kernel(const float* __restrict__ norm,
                                                   float* __restrict__ feats,
                                                   const float* __restrict__ rgam,
                                                   int H, int W, int k, int foff) {
  extern __shared__ float blk[];  // k*k
  __shared__ float red[8][26];
  __shared__ float st[25];
  __shared__ float rns[5];
  __shared__ unsigned long long bmin[5][8];

  const int tid = threadIdx.x;
  const int wave = tid >> 5, lane = tid & 31;
  const int blkid = blockIdx.x;
  const int b = blockIdx.y;
  const int bi = blkid / NBW, bj = blkid % NBW;
  const int n = k * k;
  const float* base = norm + ((size_t)b * H + (size_t)bi * k) * W + (size_t)bj * k;

  // Async bulk copy global -> LDS, 16B per lane per issue.  All addresses are
  // 16B aligned (row pitch W*4 and column offsets are multiples of 16).
  for (int idx = tid * 4; idx < n; idx += 256 * 4) {
    const float* g = base + (size_t)(idx / k) * W + (idx % k);
    unsigned lds_addr = (unsigned)(uintptr_t)&blk[idx];
    asm volatile("global_load_async_to_lds_b128 %0, %1, off"
                 :: "v"(lds_addr), "v"(g)
                 : "memory");
  }
  asm volatile("s_wait_asynccnt 0x0" ::: "memory");
  __syncthreads();

  float cl[5] = {0, 0, 0, 0, 0}, cr[5] = {0, 0, 0, 0, 0};
  float sl[5] = {0, 0, 0, 0, 0}, sr[5] = {0, 0, 0, 0, 0}, sa[5] = {0, 0, 0, 0, 0};
  for (int idx = tid; idx < n; idx += 256) {
    int i = idx / k, j = idx % k;
    int im = i ? i - 1 : k - 1;                 // circular shifts within block (jnp.roll)
    int jm = j ? j - 1 : k - 1;
    int jp = (j == k - 1) ? 0 : j + 1;
    float v = blk[idx];
    float p[5];
    p[0] = v;
    p[1] = v * blk[i * k + jm];    // roll (0, 1)
    p[2] = v * blk[im * k + j];    // roll (1, 0)
    p[3] = v * blk[im * k + jm];   // roll (1, 1)
    p[4] = v * blk[im * k + jp];   // roll (1,-1)
#pragma unroll
    for (int q = 0; q < 5; ++q) {
      float x = p[q];
      float x2 = x * x;
      if (x < 0.f)      { cl[q] += 1.f; sl[q] += x2; }
      else if (x > 0.f) { cr[q] += 1.f; sr[q] += x2; }
      sa[q] += fabsf(x);
    }
  }
#pragma unroll
  for (int q = 0; q < 5; ++q)
    for (int off = 16; off > 0; off >>= 1) {
      cl[q] += __shfl_down(cl[q], off);
      cr[q] += __shfl_down(cr[q], off);
      sl[q] += __shfl_down(sl[q], off);
      sr[q] += __shfl_down(sr[q], off);
      sa[q] += __shfl_down(sa[q], off);
    }
  if (lane == 0) {
#pragma unroll
    for (int q = 0; q < 5; ++q) {
      red[wave][q * 5 + 0] = cl[q];
      red[wave][q * 5 + 1] = cr[q];
      red[wave][q * 5 + 2] = sl[q];
      red[wave][q * 5 + 3] = sr[q];
      red[wave][q * 5 + 4] = sa[q];
    }
  }
  __syncthreads();
  if (tid < 25) {
    float s = 0.f;
#pragma unroll
    for (int w8 = 0; w8 < 8; ++w8) s += red[w8][tid];
    st[tid] = s;
  }
  __syncthreads();

  float lstd = 0.f, rstd = 0.f;
  if (tid < 5) {
    float Cl = st[tid * 5 + 0], Cr = st[tid * 5 + 1];
    float Sl = st[tid * 5 + 2], Sr = st[tid * 5 + 3], Sa = st[tid * 5 + 4];
    lstd = sqrtf(Sl / (Cl + 1e-8f));
    rstd = sqrtf(Sr / (Cr + 1e-8f));
    float g = lstd / rstd;
    float nf = (float)n;
    float rhat = (Sa * Sa) / (nf * (Sl + Sr));     // mean|x|^2 / mean(x^2)
    float g2 = g * g;
    rns[tid] = rhat * (g2 * g + 1.f) * (g + 1.f) / ((g2 + 1.f) * (g2 + 1.f));
  }
  __syncthreads();

  float rnl[5];
#pragma unroll
  for (int q = 0; q < 5; ++q) rnl[q] = rns[q];
  unsigned long long best[5];
#pragma unroll
  for (int q = 0; q < 5; ++q) best[q] = ~0ull;
  for (int i = tid; i < NG; i += 256) {
    float rg = rgam[i];
#pragma unroll
    for (int q = 0; q < 5; ++q) {
      unsigned long long key =
          ((unsigned long long)__float_as_uint(fabsf(rg - rnl[q])) << 32) | (unsigned)i;
      if (key < best[q]) best[q] = key;
    }
  }
#pragma unroll
  for (int q = 0; q < 5; ++q)
    for (int off = 16; off > 0; off >>= 1) {
      unsigned long long o = __shfl_down(best[q], off);
      if (o < best[q]) best[q] = o;
    }
  if (lane == 0) {
#pragma unroll
    for (int q = 0; q < 5; ++q) bmin[q][wave] = best[q];
  }
  __syncthreads();

  if (tid < 5) {
    unsigned long long m = bmin[tid][0];
#pragma unroll
    for (int w8 = 1; w8 < 8; ++w8)
      if (bmin[tid][w8] < m) m = bmin[tid][w8];
    int gi = (int)(unsigned)m;
    float a = 0.2f + 0.001f * (float)gi;
    float cnv = sqrtf(expf(lgammaf(1.f / a) - lgammaf(3.f / a)));
    float bl = lstd * cnv, br = rstd * cnv;
    float* F = feats + ((size_t)b * NB + blkid) * NF + foff;
    if (tid == 0) {
      F[0] = a;
      F[1] = 0.5f * (bl + br);
    } else {
      int b4 = 2 + (tid - 1) * 4;
      float mn = (br - bl) * expf(lgammaf(2.f / a) - lgammaf(1.f / a));
      F[b4 + 0] = a;
      F[b4 + 1] = mn;
      F[b4 + 2] = bl;
      F[b4 + 3] = br;
    }
  }
}

// ---------------------------------------------------------------------------
// Per-batch feature mean, diff = mu_pris - mu_dist, and zero-padded centered
// feature matrix xm[b][KPAD][FPAD] for the WMMA covariance GEMM.
// ---------------------------------------------------------------------------
__global__ __launch_bounds__(128) void mu_xm_kernel(const float* __restrict__ feats,
                                                    const float* __restrict__ mu_pris,
                                                    float* __restrict__ diffb,
                                                    float* __restrict__ xmb) {
  const int b = blockIdx.x, tid = threadIdx.x;
  __shared__ float mu[NF];
  if (tid < NF) {
    float s = 0.f;
    for (int nn = 0; nn < NB; ++nn) s += feats[((size_t)b * NB + nn) * NF + tid];
    float m = s * (1.f / (float)NB);
    mu[tid] = m;
    diffb[b * NF + tid] = mu_pris[tid] - m;
  }
  __syncthreads();
  float* X = xmb + (size_t)b * KPAD * FPAD;
  for (int idx = tid; idx < KPAD * FPAD; idx += 128) {
    int nrow = idx / FPAD, f = idx % FPAD;
    X[idx] = (nrow < NB && f < NF) ? (feats[((size_t)b * NB + nrow) * NF + f] - mu[f]) : 0.f;
  }
}

// ---------------------------------------------------------------------------
// cov = (cov_pris + X^T X / (NB-1)) / 2 via V_WMMA_F32_16X16X4_F32.
// One wave per 16x16 output tile; 31 chained f32 WMMAs over K=124.
// Layouts per ISA 7.12.2: A/B element at K = vgpr + 2*(lane>=16);
// C/D element at M = vgpr + 8*(lane>=16), N = lane&15.
// ---------------------------------------------------------------------------
__global__ __launch_bounds__(32) void cov_wmma_kernel(const float* __restrict__ xm,
                                                      const float* __restrict__ cov_pris,
                                                      float* __restrict__ cov) {
  const int ti = blockIdx.x, tj = blockIdx.y, b = blockIdx.z;
  const int lane = threadIdx.x;
  const int half = lane >> 4;
  const int l = lane & 15;
  const float* X = xm + (size_t)b * KPAD * FPAD;
  const int fA = ti * 16 + l;  // A row  (M = feature f), A = X^T
  const int gB = tj * 16 + l;  // B col  (N = feature g), B = X
  v8f acc = {};
  for (int k0 = 0; k0 < KPAD; k0 += 4) {
    v2f av, bv;
    av.x = X[(size_t)(k0 + half * 2 + 0) * FPAD + fA];
    av.y = X[(size_t)(k0 + half * 2 + 1) * FPAD + fA];
    bv.x = X[(size_t)(k0 + half * 2 + 0) * FPAD + gB];
    bv.y = X[(size_t)(k0 + half * 2 + 1) * FPAD + gB];
    acc = __builtin_amdgcn_wmma_f32_16x16x4_f32(false, av, false, bv, (short)0, acc,
                                                false, false);
  }
  const float s = 1.f / (float)(NB - 1);
#pragma unroll
  for (int r = 0; r < 8; ++r) {
    int f = ti * 16 + r + half * 8;
    int g = tj * 16 + l;
    if (f < NF && g < NF)
      cov[((size_t)b * NF + f) * NF + g] = 0.5f * (cov_pris[f * NF + g] + acc[r] * s);
  }
}

// ---------------------------------------------------------------------------
// quad = diff^T cov^{-1} diff via 36x36 Gauss-Jordan with partial pivoting
// (cov is SPD: SPD pristine + PSD sample).  out[b] = sqrt(max(quad, 0)).
// ---------------------------------------------------------------------------
__global__ __launch_bounds__(64) void solve_kernel(const float* __restrict__ cov,
                                                   const float* __restrict__ diffb,
                                                   float* __restrict__ out) {
  const int b = blockIdx.x, tid = threadIdx.x;
  __shared__ float M[36][38];
  __shared__ float dv[36];
  __shared__ int piv_s;
  for (int idx = tid; idx < 36 * 36; idx += 64) {
    int r = idx / 36, c = idx % 36;
    M[r][c] = cov[((size_t)b * 36 + r) * 36 + c];
  }
  if (tid < 36) {
    dv[tid] = diffb[b * 36 + tid];
    M[tid][36] = dv[tid];
  }
  __syncthreads();
  for (int p = 0; p < 36; ++p) {
    if (tid == 0) {
      int bestr = p;
      float bv = fabsf(M[p][p]);
      for (int r = p + 1; r < 36; ++r) {
        float v = fabsf(M[r][p]);
        if (v > bv) { bv = v; bestr = r; }
      }
      piv_s = bestr;
    }
    __syncthreads();
    int pr = piv_s;
    if (tid < 37 && pr != p) {
      float t = M[p][tid];
      M[p][tid] = M[pr][tid];
      M[pr][tid] = t;
    }
    __syncthreads();
    float pv = M[p][p];
    float inv = (fabsf(pv) > 1e-30f) ? 1.f / pv : 0.f;
    if (tid < 37) M[p][tid] *= inv;
    __syncthreads();
    if (tid < 36 && tid != p) {
      float f = M[tid][p];
      for (int c = p; c < 37; ++c) M[tid][c] -= f * M[p][c];
    }
    __syncthreads();
  }
  if (tid == 0) {
    float q = 0.f;
    for (int r = 0; r < 36; ++r) q += dv[r] * M[r][36];
    out[b] = sqrtf(fmaxf(q, 0.f));
  }
}

// ---------------------------------------------------------------------------
extern "C" void kernel_launch(void* const* d_in, const int* in_sizes, int n_in,
                              void* d_out, int out_size, void* d_ws, size_t ws_size,
                              hipStream_t stream) {
  const float* image    = (const float*)d_in[0];  // (16,1,1056,1056)
  const float* mu_pris  = (const float*)d_in[1];  // (36,)
  const float* cov_pris = (const float*)d_in[2];  // (36,36)
  float* out = (float*)d_out;                     // (16,)
  (void)in_sizes; (void)n_in; (void)out_size; (void)ws_size;

  char* w = (char*)d_ws;
  size_t off = 0;
  auto take = [&](size_t elems) -> float* {
    float* p = (float*)(w + off);
    off += (elems * sizeof(float) + 255) & ~(size_t)255;
    return p;
  };
  float* norm1 = take((size_t)BATCH * H1 * W1);   // 71.4 MB
  float* ds    = take((size_t)BATCH * H2 * W2);   // 17.8 MB
  float* norm2 = take((size_t)BATCH * H2 * W2);   // 17.8 MB
  float* feats = take((size_t)BATCH * NB * NF);
  float* rgam  = take(NG);
  float* diffb = take((size_t)BATCH * NF);
  float* xmb   = take((size_t)BATCH * KPAD * FPAD);
  float* covb  = take((size_t)BATCH * NF * NF);

  rgam_init_kernel<<<(NG + 255) / 256, 256, 0, stream>>>(rgam);
  mscn_kernel<<<dim3(W1 / 16, H1 / 16, BATCH), 256, 0, stream>>>(image, norm1, H1, W1);
  {
    int total = BATCH * H2 * W2;
    downsample_kernel<<<(total + 255) / 256, 256, 0, stream>>>(image, ds);
  }
  mscn_kernel<<<dim3(W2 / 16, H2 / 16, BATCH), 256, 0, stream>>>(ds, norm2, H2, W2);
  feat_kernel<<<dim3(NB, BATCH), 256, 96 * 96 * sizeof(float), stream>>>(
      norm1, feats, rgam, H1, W1, 96, 0);
  feat_kernel<<<dim3(NB, BATCH), 256, 48 * 48 * sizeof(float), stream>>>(
      norm2, feats, rgam, H2, W2, 48, 18);
  mu_xm_kernel<<<BATCH, 128, 0, stream>>>(feats, mu_pris, diffb, xmb);
  cov_wmma_kernel<<<dim3(3, 3, BATCH), 32, 0, stream>>>(xmb, cov_pris, covb);
  solve_kernel<<<BATCH, 64, 0, stream>>>(covb, diffb, out);
}